// DPSR_44856638440090
// MI455X (gfx1250) — compile-verified
//
#include <hip/hip_runtime.h>
#include <hip/hip_bf16.h>
#include <math.h>

typedef __attribute__((ext_vector_type(2))) float v2f;
typedef __attribute__((ext_vector_type(8))) float v8f;

#define NFFT   128
#define NG     (128*128*128)          // 2,097,152 grid cells
#define NPTS   131072
#define SIGMA  5.0f
#define M_IND  0.5f
#define EPS_K  1e-6f
#define TWO_PI 6.28318530717958647692f

// ---------------------------------------------------------------------------
// 1) Trilinear rasterization of normals onto [3][128][128][128] (atomic add)
// ---------------------------------------------------------------------------
__global__ void dpsr_rasterize(const float* __restrict__ pts,
                               const float* __restrict__ nrm,
                               float* __restrict__ V, int n)
{
    int p = blockIdx.x * blockDim.x + threadIdx.x;
    if (p >= n) return;
    float tx = pts[p*3+0] * (float)NFFT;
    float ty = pts[p*3+1] * (float)NFFT;
    float tz = pts[p*3+2] * (float)NFFT;
    int lx = (int)floorf(tx), ly = (int)floorf(ty), lz = (int)floorf(tz);
    int hx = ((int)ceilf(tx)) & (NFFT-1);
    int hy = ((int)ceilf(ty)) & (NFFT-1);
    int hz = ((int)ceilf(tz)) & (NFFT-1);
    float fx = tx - (float)lx, fy = ty - (float)ly, fz = tz - (float)lz;
    float n0 = nrm[p*3+0], n1 = nrm[p*3+1], n2 = nrm[p*3+2];
    #pragma unroll
    for (int c = 0; c < 8; ++c) {
        int cx = (c >> 2) & 1, cy = (c >> 1) & 1, cz = c & 1;
        int ix = cx ? hx : lx;
        int iy = cy ? hy : ly;
        int iz = cz ? hz : lz;
        float w = (cx ? fx : 1.f - fx) * (cy ? fy : 1.f - fy) * (cz ? fz : 1.f - fz);
        int flat = (ix * NFFT + iy) * NFFT + iz;
        atomicAdd(&V[0*NG + flat], w * n0);
        atomicAdd(&V[1*NG + flat], w * n1);
        atomicAdd(&V[2*NG + flat], w * n2);
    }
}

// ---------------------------------------------------------------------------
// 2) Batched length-128 DFT along one axis as WMMA f32 16x16x4 matmul.
//    Element (l0,k,l1) lives at ((l0*128+k) << sh) + l1  with  L1 = 1<<sh.
//    One block = 16 lines, 128 threads = 4 waves, each wave -> 2 col tiles.
//    sign = -1 forward, +1 inverse.  Xim==nullptr treated as zero.
// ---------------------------------------------------------------------------
__global__ __launch_bounds__(128)
void dpsr_dft(const float* __restrict__ Xre, const float* __restrict__ Xim,
              float* __restrict__ Yre, float* __restrict__ Yim,
              int sh, float sgn, float scale)
{
    __shared__ float Are[16][132];
    __shared__ float Aim[16][132];

    const int L1   = 1 << sh;
    const int mask = L1 - 1;
    const int lid0 = blockIdx.x * 16;

    // stage 16 lines x 128 taps into LDS (coalesced for both layouts)
    for (int i = threadIdx.x; i < 16 * NFFT; i += 128) {
        int line, k;
        if (sh == 0) { line = i >> 7;  k = i & 127; }
        else         { line = i & 15;  k = i >> 4;  }
        int    lid = lid0 + line;
        int    l0  = lid >> sh;
        int    l1  = lid & mask;
        size_t off = ((size_t)(l0 * NFFT + k) << sh) + (size_t)l1;
        Are[line][k] = Xre[off];
        Aim[line][k] = Xim ? Xim[off] : 0.0f;
    }
    __syncthreads();

    const int wave  = threadIdx.x >> 5;
    const int lane  = threadIdx.x & 31;
    const int mrow  = lane & 15;            // A-fragment row  (M)
    const int khalf = (lane >> 4) << 1;     // A/B-fragment K half: 0 or 2
    const float step = TWO_PI / (float)NFFT;

    for (int t = 0; t < 2; ++t) {
        const int n0   = (wave * 2 + t) * 16;
        const int ncol = n0 + (lane & 15); // B/D column (N)
        v8f cre = {};
        v8f cim = {};

        #pragma unroll 4
        for (int kb = 0; kb < NFFT; kb += 4) {
            // A fragments (16x4) from LDS
            v2f are = *(const v2f*)&Are[mrow][kb + khalf];
            v2f aim = *(const v2f*)&Aim[mrow][kb + khalf];
            // B fragments (4x16): twiddles, exact phase mod 128
            int kk0 = kb + khalf;
            int p0  = (kk0 * ncol) & (NFFT - 1);
            int p1  = ((kk0 + 1) * ncol) & (NFFT - 1);
            float s0, c0, s1, c1;
            __sincosf((float)p0 * step, &s0, &c0);
            __sincosf((float)p1 * step, &s1, &c1);
            v2f bre  = {c0, c1};
            v2f bim  = {sgn * s0, sgn * s1};
            v2f bimn = {-bim.x, -bim.y};
            // complex matmul: Yre += Are*Wre - Aim*Wim ; Yim += Are*Wim + Aim*Wre
            cre = __builtin_amdgcn_wmma_f32_16x16x4_f32(false, are, false, bre,  (short)0, cre, false, false);
            cre = __builtin_amdgcn_wmma_f32_16x16x4_f32(false, aim, false, bimn, (short)0, cre, false, false);
            cim = __builtin_amdgcn_wmma_f32_16x16x4_f32(false, are, false, bim,  (short)0, cim, false, false);
            cim = __builtin_amdgcn_wmma_f32_16x16x4_f32(false, aim, false, bre,  (short)0, cim, false, false);
        }

        // D layout: vgpr v holds row m = v + (lane<16 ? 0 : 8), col = ncol
        const int mbase = (lane < 16) ? 0 : 8;
        #pragma unroll
        for (int v = 0; v < 8; ++v) {
            int    m   = mbase + v;
            int    lid = lid0 + m;
            int    l0  = lid >> sh;
            int    l1  = lid & mask;
            size_t off = ((size_t)(l0 * NFFT + ncol) << sh) + (size_t)l1;
            Yre[off] = cre[v] * scale;
            Yim[off] = cim[v] * scale;
        }
    }
}

// ---------------------------------------------------------------------------
// 3) Spectral filter: chi_tilde = g * (-i/2pi) * sum(u * v_tilde) / (u2+eps)
// ---------------------------------------------------------------------------
__global__ void dpsr_spectral(const float* __restrict__ Vre,
                              const float* __restrict__ Vim,
                              float* __restrict__ Cre,
                              float* __restrict__ Cim)
{
    int idx = blockIdx.x * blockDim.x + threadIdx.x;
    if (idx >= NG) return;
    int i = idx >> 14;
    int j = (idx >> 7) & 127;
    int k = idx & 127;
    float fx = (float)(i < 64 ? i : i - 128);
    float fy = (float)(j < 64 ? j : j - 128);
    float fz = (float)(k < 64 ? k : k - 128);
    float u2 = fx*fx + fy*fy + fz*fz;
    float sg = SIGMA / (float)NFFT;
    float g  = __expf(-2.0f * sg * sg * u2);
    float vr = fx*Vre[idx] + fy*Vre[idx + NG] + fz*Vre[idx + 2*NG];
    float vi = fx*Vim[idx] + fy*Vim[idx + NG] + fz*Vim[idx + 2*NG];
    float inv = 1.0f / (u2 + EPS_K);
    vr *= inv; vi *= inv;
    float f = g / TWO_PI;                 // (-i/2pi)*(vr+i*vi) = (vi - i*vr)/2pi
    Cre[idx] =  f * vi;
    Cim[idx] = -f * vr;
}

// ---------------------------------------------------------------------------
// 4) Trilinear gather at the points + global mean accumulation
// ---------------------------------------------------------------------------
__global__ void dpsr_interp_mean(const float* __restrict__ chi,
                                 const float* __restrict__ pts,
                                 int n, float* __restrict__ accum)
{
    __shared__ float red[256];
    int p = blockIdx.x * blockDim.x + threadIdx.x;
    float val = 0.0f;
    if (p < n) {
        float tx = pts[p*3+0] * (float)NFFT;
        float ty = pts[p*3+1] * (float)NFFT;
        float tz = pts[p*3+2] * (float)NFFT;
        int lx = (int)floorf(tx), ly = (int)floorf(ty), lz = (int)floorf(tz);
        int hx = ((int)ceilf(tx)) & (NFFT-1);
        int hy = ((int)ceilf(ty)) & (NFFT-1);
        int hz = ((int)ceilf(tz)) & (NFFT-1);
        float fx = tx - (float)lx, fy = ty - (float)ly, fz = tz - (float)lz;
        #pragma unroll
        for (int c = 0; c < 8; ++c) {
            int cx = (c >> 2) & 1, cy = (c >> 1) & 1, cz = c & 1;
            int ix = cx ? hx : lx;
            int iy = cy ? hy : ly;
            int iz = cz ? hz : lz;
            float w = (cx ? fx : 1.f-fx) * (cy ? fy : 1.f-fy) * (cz ? fz : 1.f-fz);
            val += w * chi[(ix * NFFT + iy) * NFFT + iz];
        }
    }
    red[threadIdx.x] = val;
    __syncthreads();
    for (int s = 128; s > 0; s >>= 1) {
        if (threadIdx.x < s) red[threadIdx.x] += red[threadIdx.x + s];
        __syncthreads();
    }
    if (threadIdx.x == 0) atomicAdd(accum, red[0]);
}

// ---------------------------------------------------------------------------
// 5) Center, normalize by |chi0|, scale by M_IND
// ---------------------------------------------------------------------------
__global__ void dpsr_finalize(const float* __restrict__ chi,
                              const float* __restrict__ accum,
                              float* __restrict__ out, int n_pts)
{
    int idx = blockIdx.x * blockDim.x + threadIdx.x;
    if (idx >= NG) return;
    float mean = accum[0] / (float)n_pts;
    float chi0 = chi[0] - mean;
    float s    = M_IND / fabsf(chi0);
    out[idx]   = s * (chi[idx] - mean);
}

// ---------------------------------------------------------------------------
// Launcher
// ---------------------------------------------------------------------------
extern "C" void kernel_launch(void* const* d_in, const int* in_sizes, int n_in,
                              void* d_out, int out_size, void* d_ws, size_t ws_size,
                              hipStream_t stream)
{
    const float* pts = (const float*)d_in[0];   // [1, 131072, 3]
    const float* nrm = (const float*)d_in[1];   // [1, 131072, 3]
    float*       out = (float*)d_out;           // [1,128,128,128]

    float* ws  = (float*)d_ws;
    float* Are = ws;                    // 3*NG
    float* Aim = ws + (size_t)3*NG;     // 3*NG
    float* Bre = ws + (size_t)6*NG;     // 3*NG
    float* Bim = ws + (size_t)9*NG;     // 3*NG
    float* acc = ws + (size_t)12*NG;    // 1

    // zero rasterization target + mean accumulator
    hipMemsetAsync(Are, 0, (size_t)3*NG*sizeof(float), stream);
    hipMemsetAsync(acc, 0, sizeof(float), stream);

    // 1) rasterize normals -> real field v in Are[3][128^3]
    dpsr_rasterize<<<(NPTS + 255)/256, 256, 0, stream>>>(pts, nrm, Are, NPTS);

    // 2) forward 3D FFT (3 channels): z, y, x axes; 49152 lines -> 3072 blocks
    dpsr_dft<<<3072, 128, 0, stream>>>(Are, nullptr, Bre, Bim, /*sh=*/0,  -1.0f, 1.0f);
    dpsr_dft<<<3072, 128, 0, stream>>>(Bre, Bim,     Are, Aim, /*sh=*/7,  -1.0f, 1.0f);
    dpsr_dft<<<3072, 128, 0, stream>>>(Are, Aim,     Bre, Bim, /*sh=*/14, -1.0f, 1.0f);

    // 3) spectral filter: V~ (B, 3ch) -> chi~ (A, ch0)
    dpsr_spectral<<<(NG + 255)/256, 256, 0, stream>>>(Bre, Bim, Are, Aim);

    // 4) inverse 3D FFT (1 channel): 16384 lines -> 1024 blocks, scale 1/128/axis
    const float is = 1.0f / (float)NFFT;
    dpsr_dft<<<1024, 128, 0, stream>>>(Are, Aim, Bre, Bim, /*sh=*/0,  1.0f, is);
    dpsr_dft<<<1024, 128, 0, stream>>>(Bre, Bim, Are, Aim, /*sh=*/7,  1.0f, is);
    dpsr_dft<<<1024, 128, 0, stream>>>(Are, Aim, Bre, Bim, /*sh=*/14, 1.0f, is);
    // chi_prime (real part) = Bre[0..NG)

    // 5) gather at points -> mean; then center/normalize
    dpsr_interp_mean<<<(NPTS + 255)/256, 256, 0, stream>>>(Bre, pts, NPTS, acc);
    dpsr_finalize<<<(NG + 255)/256, 256, 0, stream>>>(Bre, acc, out, NPTS);
}